// QuantizationModifier_51719996178490
// MI455X (gfx1250) — compile-verified
//
#include <hip/hip_runtime.h>
#include <math.h>

typedef float vf4 __attribute__((ext_vector_type(4)));
typedef int   vi4 __attribute__((ext_vector_type(4)));

#define BT 256           // threads per block (8 wave32 waves)
#define NB_MAX 2048      // partial-reduction blocks
#define QB 4096          // quantize-pass blocks

// ---- gfx1250 async global->LDS path (guarded; falls back to direct loads) ----
#if defined(__AMDGCN__) && __has_builtin(__builtin_amdgcn_global_load_async_to_lds_b128)
#define HAVE_ASYNC_LDS 1
#else
#define HAVE_ASYNC_LDS 0
#endif

#if defined(__AMDGCN__) && __has_builtin(__builtin_amdgcn_s_wait_asynccnt)
#define WAIT_ASYNC(n) __builtin_amdgcn_s_wait_asynccnt(n)
#elif defined(__AMDGCN__)
#define WAIT_ASYNC(n) asm volatile("s_wait_asynccnt " #n ::: "memory")
#else
#define WAIT_ASYNC(n)
#endif

__device__ __forceinline__ float min4(vf4 v) {
    return fminf(fminf(v.x, v.y), fminf(v.z, v.w));
}
__device__ __forceinline__ float max4(vf4 v) {
    return fmaxf(fmaxf(v.x, v.y), fmaxf(v.z, v.w));
}

// ---------------- Pass 1: per-block min/max partials ----------------
// stride/fullIters computed host-side: no 64-bit scalar division in the kernel.
__global__ __launch_bounds__(BT) void k_minmax_partial(const vf4* __restrict__ x4,
                                                       int n4, int stride, int fullIters,
                                                       int nTail,
                                                       const float* __restrict__ xTail,
                                                       float* __restrict__ partial) {
    const int g0 = (int)blockIdx.x * BT + (int)threadIdx.x;
    float vmin = INFINITY, vmax = -INFINITY;

#if HAVE_ASYNC_LDS
    __shared__ vf4 tile[2][BT];
    auto gbase = (__attribute__((address_space(1))) char*)x4;
    auto lds0  = (__attribute__((address_space(3))) vi4*)&tile[0][threadIdx.x];
    auto lds1  = (__attribute__((address_space(3))) vi4*)&tile[1][threadIdx.x];
    if (fullIters > 0) {
        __builtin_amdgcn_global_load_async_to_lds_b128(
            (__attribute__((address_space(1))) vi4*)(gbase + (size_t)g0 * 16), lds0, 0, 0);
    }
    for (int k = 0; k < fullIters; ++k) {
        if (k + 1 < fullIters) {
            // prefetch next tile into the other buffer, wait only for the older group
            __builtin_amdgcn_global_load_async_to_lds_b128(
                (__attribute__((address_space(1))) vi4*)(gbase + (size_t)(g0 + (k + 1) * stride) * 16),
                (k & 1) ? lds0 : lds1, 0, 0);
            WAIT_ASYNC(1);
        } else {
            WAIT_ASYNC(0);
        }
        vf4 v = tile[k & 1][threadIdx.x];   // each lane reads only its own staged slot
        vmin = fminf(vmin, min4(v));
        vmax = fmaxf(vmax, max4(v));
    }
    for (int i = g0 + fullIters * stride; i < n4; i += stride) {
        vf4 v = x4[i];
        vmin = fminf(vmin, min4(v));
        vmax = fmaxf(vmax, max4(v));
    }
#else
    (void)fullIters;
    for (int i = g0; i < n4; i += stride) {
        vf4 v = x4[i];
        vmin = fminf(vmin, min4(v));
        vmax = fmaxf(vmax, max4(v));
    }
#endif
    // scalar tail (n not multiple of 4) — covered by block 0
    if (blockIdx.x == 0 && (int)threadIdx.x < nTail) {
        float t = xTail[threadIdx.x];
        vmin = fminf(vmin, t);
        vmax = fmaxf(vmax, t);
    }

    __shared__ float smin[BT], smax[BT];
    smin[threadIdx.x] = vmin;
    smax[threadIdx.x] = vmax;
    __syncthreads();
    for (int s = BT / 2; s > 0; s >>= 1) {
        if ((int)threadIdx.x < s) {
            smin[threadIdx.x] = fminf(smin[threadIdx.x], smin[threadIdx.x + s]);
            smax[threadIdx.x] = fmaxf(smax[threadIdx.x], smax[threadIdx.x + s]);
        }
        __syncthreads();
    }
    if (threadIdx.x == 0) {
        partial[2 * blockIdx.x]     = smin[0];
        partial[2 * blockIdx.x + 1] = smax[0];
    }
}

// ---------------- Pass 2: reduce partials -> {min, step} ----------------
__global__ __launch_bounds__(BT) void k_minmax_final(const float* __restrict__ partial,
                                                     int nb, float* __restrict__ params) {
    float vmin = INFINITY, vmax = -INFINITY;
    for (int i = threadIdx.x; i < nb; i += BT) {
        vmin = fminf(vmin, partial[2 * i]);
        vmax = fmaxf(vmax, partial[2 * i + 1]);
    }
    __shared__ float smin[BT], smax[BT];
    smin[threadIdx.x] = vmin;
    smax[threadIdx.x] = vmax;
    __syncthreads();
    for (int s = BT / 2; s > 0; s >>= 1) {
        if ((int)threadIdx.x < s) {
            smin[threadIdx.x] = fminf(smin[threadIdx.x], smin[threadIdx.x + s]);
            smax[threadIdx.x] = fmaxf(smax[threadIdx.x], smax[threadIdx.x + s]);
        }
        __syncthreads();
    }
    if (threadIdx.x == 0) {
        float gmin = smin[0];
        float gmax = smax[0];
        params[0] = gmin;
        params[1] = (gmax - gmin) / 256.0f;   // step, matches reference
    }
}

// ---------------- Pass 3: quantize (L2-resident reads, NT writes) ----------------
__device__ __forceinline__ float qone(float x, float gmin, float step) {
    float idx = floorf((x - gmin) / step);            // exact IEEE div, same as reference
    idx = fminf(fmaxf(idx, 0.0f), 255.0f);            // clip to [0, 2^8-1]
    return gmin + (idx + 0.5f) * step;
}

__global__ __launch_bounds__(BT) void k_quantize(const vf4* __restrict__ x4,
                                                 vf4* __restrict__ o4,
                                                 const float* __restrict__ params,
                                                 int n4, int nTail,
                                                 const float* __restrict__ xTail,
                                                 float* __restrict__ oTail) {
    const float gmin = params[0];
    const float step = params[1];
    const int stride = QB * BT;
    for (int i = (int)blockIdx.x * BT + (int)threadIdx.x; i < n4; i += stride) {
        if (i + stride < n4)
            __builtin_prefetch((const char*)x4 + (size_t)(i + stride) * 16, 0, 1);  // global_prefetch_b8
        vf4 v = x4[i];
        vf4 r;
        r.x = qone(v.x, gmin, step);
        r.y = qone(v.y, gmin, step);
        r.z = qone(v.z, gmin, step);
        r.w = qone(v.w, gmin, step);
        __builtin_nontemporal_store(r, &o4[i]);       // NT: keep x resident in L2
    }
    if (blockIdx.x == 0 && (int)threadIdx.x < nTail)
        oTail[threadIdx.x] = qone(xTail[threadIdx.x], gmin, step);
}

extern "C" void kernel_launch(void* const* d_in, const int* in_sizes, int n_in,
                              void* d_out, int out_size, void* d_ws, size_t ws_size,
                              hipStream_t stream) {
    const float* x = (const float*)d_in[0];
    float* out = (float*)d_out;
    const int n     = in_sizes[0];   // 64*512*1024 = 33,554,432
    const int n4    = n >> 2;
    const int nTail = n & 3;

    // ws layout (floats): [0 .. 2*nb) partial {min,max} pairs; [2*nb .. 2*nb+2) = {min, step}
    float* wsf = (float*)d_ws;
    int nb = NB_MAX;
    const long cap = (long)(ws_size / sizeof(float));
    if (cap < 2L * nb + 8) nb = (int)((cap - 8) / 2);
    if (nb < 1) nb = 1;
    float* partial = wsf;
    float* params  = wsf + 2L * nb;

    const float* xTail = x + (size_t)n4 * 4;
    float* oTail = out + (size_t)n4 * 4;

    const int stride1   = nb * BT;
    const int fullIters = n4 / stride1;   // host-side: no 64-bit division in the kernel

    k_minmax_partial<<<nb, BT, 0, stream>>>((const vf4*)x, n4, stride1, fullIters,
                                            nTail, xTail, partial);
    k_minmax_final<<<1, BT, 0, stream>>>(partial, nb, params);
    k_quantize<<<QB, BT, 0, stream>>>((const vf4*)x, (vf4*)out, params, n4, nTail, xTail, oTail);
}